// SpatialAttention_22686017257910
// MI455X (gfx1250) — compile-verified
//
#include <hip/hip_runtime.h>
#include <hip/hip_bf16.h>

// SpatialAttention: x[16,64,512,512] f32 -> channel max/mean -> 7x7 conv (w[1,2,7,7]) -> sigmoid
// Pass 1: bandwidth-bound channel reduction (1 GiB NT stream), writes {max,avg} float2 pairs to ws (32 MiB, L2-resident).
// Pass 2: conv as per-wave f32 WMMA (V_WMMA_F32_16X16X4_F32): M=16 pixels, K=(tap,channel), 49 chained WMMAs.

typedef __attribute__((ext_vector_type(2))) float v2f;
typedef __attribute__((ext_vector_type(4))) float v4f;
typedef __attribute__((ext_vector_type(8))) float v8f;

#define HW   262144   // 512*512
#define NCH  64
#define IMW  512
#define TW   22       // tile width  (16 + 6 halo)
#define TN   154      // 7 * 22 tile entries

// ---------------- Pass 1: channel max/mean reduction ----------------
// grid: 4096 blocks * 256 threads, each thread owns 4 consecutive pixels (b128 loads).
__global__ __launch_bounds__(256) void sa_reduce(const float* __restrict__ x,
                                                 v2f* __restrict__ ws2) {
  const int tid = blockIdx.x * 256 + threadIdx.x;   // 0 .. 1048575
  const int b   = tid >> 16;                        // HW/4 = 65536 quads per batch
  const int q   = tid & 65535;
  const v4f* p = reinterpret_cast<const v4f*>(x) + (size_t)b * (NCH * (HW / 4)) + q;

  v4f t  = __builtin_nontemporal_load(p);           // NT: don't pollute L2 with the 1 GiB stream
  v4f mx = t, sm = t;
#pragma unroll 4
  for (int c = 1; c < NCH; ++c) {
    if (c + 8 < NCH)
      __builtin_prefetch(p + (size_t)(c + 8) * (HW / 4), 0, 0);   // global_prefetch_b8
    v4f u = __builtin_nontemporal_load(p + (size_t)c * (HW / 4));
    mx.x = fmaxf(mx.x, u.x); mx.y = fmaxf(mx.y, u.y);
    mx.z = fmaxf(mx.z, u.z); mx.w = fmaxf(mx.w, u.w);
    sm += u;
  }
  const float s = 1.0f / 64.0f;
  v4f o0 = { mx.x, sm.x * s, mx.y, sm.y * s };
  v4f o1 = { mx.z, sm.z * s, mx.w, sm.w * s };
  v4f* dst = reinterpret_cast<v4f*>(ws2 + (size_t)b * HW + (size_t)q * 4);
  dst[0] = o0;
  dst[1] = o1;
}

// ---------------- Pass 2: 7x7 conv + sigmoid via V_WMMA_F32_16X16X4_F32 ----------------
// Each wave computes 16 consecutive output pixels (one strip).
// GEMM view: D = A(16x4) x B(4x16) chained over 49 K-slabs (K index = 2*tap + channel):
//   A 16x4 f32 layout: lane<16 -> {K0,K1} of row m, lane>=16 -> {K2,K3}  == one float2 {max,avg} per lane.
//   B[k][n] = wflat[k] for all n  -> per-lane float2 {w_max[t], w_avg[t]}, t = 2j + (lane>=16).
__global__ __launch_bounds__(256) void sa_conv(const v2f* __restrict__ ws2,
                                               const float* __restrict__ wgt,
                                               float* __restrict__ out) {
  __shared__ v2f  s_w[64];          // wpair[t] = { w[0][t], w[1][t] }, t = ky*7+kx
  __shared__ v2f  s_tile[8][TN];    // per-wave 7x22 halo tile of {max,avg}
  __shared__ float s_d[8][256];     // per-wave 16x16 D dump

  const int tid  = threadIdx.x;
  const int wv   = tid >> 5;
  const int lane = tid & 31;

  if (tid < 49) {
    v2f wp = { wgt[tid], wgt[49 + tid] };
    s_w[tid] = wp;
  }

  const int strip = blockIdx.x * 8 + wv;      // 262144 strips total
  const int b     = strip >> 14;              // 512 rows * 32 strips/row = 16384 per batch
  const int rem   = strip & 16383;
  const int y     = rem >> 5;
  const int x0    = (rem & 31) << 4;
  const v2f* src  = ws2 + (size_t)b * HW;

  // Cooperative (per-wave) halo tile load with zero padding; non-divergent (clamp + select).
  for (int i = lane; i < TN; i += 32) {
    int ry = i / TW;
    int rx = i - ry * TW;
    int gy = y - 3 + ry;
    int gx = x0 - 3 + rx;
    bool inb = (gy >= 0) && (gy < IMW) && (gx >= 0) && (gx < IMW);
    int cy = gy < 0 ? 0 : (gy > IMW - 1 ? IMW - 1 : gy);
    int cx = gx < 0 ? 0 : (gx > IMW - 1 ? IMW - 1 : gx);
    v2f v = src[(size_t)cy * IMW + cx];
    if (!inb) { v.x = 0.0f; v.y = 0.0f; }
    s_tile[wv][i] = v;
  }
  __syncthreads();

  const int m    = lane & 15;
  const int half = lane >> 4;

  v8f acc = {0.f, 0.f, 0.f, 0.f, 0.f, 0.f, 0.f, 0.f};
#pragma unroll
  for (int j = 0; j < 49; ++j) {
    const int t0 = 2 * j, t1 = 2 * j + 1;                 // two spatial taps this slab
    const int o0 = (t0 / 7) * TW + (t0 % 7);              // compile-time after unroll
    const int o1 = (t1 / 7) * TW + (t1 % 7);
    v2f a  = s_tile[wv][(half ? o1 : o0) + m];            // A fragment: {max,avg} of (pixel m, tap)
    v2f bb = s_w[t0 + half];                              // B fragment: {w_max, w_avg} of tap
    acc = __builtin_amdgcn_wmma_f32_16x16x4_f32(
        /*neg_a=*/false, a, /*neg_b=*/false, bb,
        /*c_mod=*/(short)0, acc, /*reuse_a=*/false, /*reuse_b=*/false);
  }

  // D columns are identical; dump to LDS (same-wave DS ops are in-order), read column 0.
#pragma unroll
  for (int v = 0; v < 8; ++v)
    s_d[wv][(v + 8 * half) * 16 + m] = acc[v];

  if (lane < 16) {
    float r = s_d[wv][lane * 16];
    r = 1.0f / (1.0f + __expf(-r));
    out[(size_t)b * HW + (size_t)y * IMW + x0 + lane] = r;
  }
}

extern "C" void kernel_launch(void* const* d_in, const int* in_sizes, int n_in,
                              void* d_out, int out_size, void* d_ws, size_t ws_size,
                              hipStream_t stream) {
  const float* x = (const float*)d_in[0];   // [16,64,512,512]
  const float* w = (const float*)d_in[1];   // [1,2,7,7]
  float* out = (float*)d_out;               // [16,1,512,512]
  v2f* ws2 = (v2f*)d_ws;                    // needs 16*512*512 * 8B = 32 MiB

  sa_reduce<<<4096, 256, 0, stream>>>(x, ws2);
  sa_conv<<<32768, 256, 0, stream>>>(ws2, w, out);
}